// F0Collisions_88467736363223
// MI455X (gfx1250) — compile-verified
//
#include <hip/hip_runtime.h>
#include <math.h>

#define NXX 4096
#define NVV 1024
#define DVF 0.0078125f   // 8.0 / 1024

typedef __attribute__((ext_vector_type(2))) float v2f;
typedef __attribute__((ext_vector_type(8))) float v8f;

typedef __attribute__((address_space(1))) int as1_int;
typedef __attribute__((address_space(3))) int as3_int;

// ---------------------------------------------------------------------------
// Async LDS staging helpers (gfx1250 GLOBAL_LOAD_ASYNC_TO_LDS_B32 /
// GLOBAL_STORE_ASYNC_FROM_LDS_B32, tracked with ASYNCcnt). Guarded with
// __has_builtin so the file still compiles (synchronous fallback) if the
// toolchain lacks them.
// ---------------------------------------------------------------------------
#if defined(__has_builtin)
#if __has_builtin(__builtin_amdgcn_global_load_async_to_lds_b32) && \
    __has_builtin(__builtin_amdgcn_global_store_async_from_lds_b32)
#define HAS_ASYNC_LDS 1
#endif
#endif
#ifndef HAS_ASYNC_LDS
#define HAS_ASYNC_LDS 0
#endif

__device__ __forceinline__ void async_ld32(const float* g, float* l)
{
#if HAS_ASYNC_LDS
    __builtin_amdgcn_global_load_async_to_lds_b32(
        (as1_int*)g, (as3_int*)l, 0, 0);
#else
    *l = *g;
#endif
}

__device__ __forceinline__ void async_st32(float* g, const float* l)
{
#if HAS_ASYNC_LDS
    __builtin_amdgcn_global_store_async_from_lds_b32(
        (as1_int*)g, (as3_int*)l, 0, 0);
#else
    *g = *l;
#endif
}

__device__ __forceinline__ void wait_async0()
{
#if HAS_ASYNC_LDS
#if __has_builtin(__builtin_amdgcn_s_wait_asynccnt)
    __builtin_amdgcn_s_wait_asynccnt(0);
#else
    asm volatile("s_wait_asynccnt 0" ::: "memory");
#endif
#endif
}

// ---------------------------------------------------------------------------
// Kernel A: per-row moments via f32 WMMA.
// One wave handles 16 rows. A = f tile (16 x 4), B = [v^2, v^4, 0...] (4 x 16),
// accumulate C (16x16 f32) over K = 1024. beta = 1.5 * n2 / n4 (dv cancels).
// ---------------------------------------------------------------------------
__global__ __launch_bounds__(32) void moments_beta_kernel(
    const float* __restrict__ f, float* __restrict__ beta_out)
{
    const int lane = threadIdx.x;        // 0..31, full wave: EXEC all ones
    const int half = lane >> 4;          // 0 or 1
    const int m    = lane & 15;          // row (for A) / column N (for B)
    const int row0 = blockIdx.x * 16;

    // branchless column selectors for B: col0 = v^2, col1 = v^4, rest 0
    const float sel0 = (m == 0) ? 1.0f : 0.0f;
    const float sel1 = (m == 1) ? 1.0f : 0.0f;

    const float* rowp = f + (size_t)(row0 + m) * NVV;

    v8f acc = {};
    for (int k0 = 0; k0 < NVV; k0 += 4) {
        const int kk = k0 + 2 * half;
        v2f a = *(const v2f*)(rowp + kk);            // A: K=kk, kk+1 (8B aligned)
        const float v0 = (kk + 0.5f) * DVF;
        const float v1 = (kk + 1.5f) * DVF;
        const float v0s = v0 * v0, v1s = v1 * v1;
        v2f b;
        b.x = sel0 * v0s + sel1 * (v0s * v0s);
        b.y = sel0 * v1s + sel1 * (v1s * v1s);
        acc = __builtin_amdgcn_wmma_f32_16x16x4_f32(
            false, a, false, b, (short)0, acc, false, false);
    }

    // C layout: VGPR r -> M = r + 8*half (lanes>=16), N = lane&15
    __shared__ float mom[16][2];
    if (m < 2) {
        const int Mbase = half * 8;
#pragma unroll
        for (int r = 0; r < 8; ++r) mom[Mbase + r][m] = acc[r];
    }
    __syncthreads();
    if (lane < 16) {
        beta_out[row0 + lane] = 1.5f * mom[lane][0] / mom[lane][1];
    }
}

// ---------------------------------------------------------------------------
// Kernel B: Chang-Cooper coefficients on the fly + batched Thomas solve.
// Lane <-> row mapping; each thread owns RPT=2 rows (independent chains for
// ILP on the serial recurrence). All global traffic coalesced:
//   - f / y / c / x staged through padded LDS tiles (stride 33) using
//     ASYNCcnt-tracked global<->LDS transfers (no VGPR round trip)
//   - c stored transposed: c_ws[j*NX + row]
// Forward parks y in d_out (overwritten by x in the backward sweep).
// ---------------------------------------------------------------------------
#define RPT      2
#define BTHREADS 64
#define ROWS     (BTHREADS * RPT)   // 128 rows per block
#define TJ       32

__global__ __launch_bounds__(BTHREADS) void thomas_kernel(
    const float* __restrict__ f, const float* __restrict__ beta_arr,
    const float* __restrict__ dtp, float* __restrict__ c_ws,
    float* __restrict__ out)
{
    __shared__ float tile [ROWS][TJ + 1];
    __shared__ float ytile[ROWS][TJ + 1];
    __shared__ float ctile[ROWS][TJ + 1];

    const int tid  = threadIdx.x;
    const int lane = tid & 31;
    const int wrp  = tid >> 5;              // 0..1
    const int row0 = blockIdx.x * ROWS;
    const float dt = dtp[0];                // nu input unused (NUEE=1 in ref)

    int   rowq[RPT];
    float DoDV[RPT], tbdv[RPT];
    float a_cur[RPT], b_cur[RPT], c_prev[RPT], y_prev[RPT];
#pragma unroll
    for (int q = 0; q < RPT; ++q) {
        rowq[q] = row0 + q * BTHREADS + tid;
        const float beta = beta_arr[rowq[q]];
        DoDV[q] = 0.5f / (beta * DVF);      // D / dv
        tbdv[q] = 2.0f * beta * DVF;        // w = 2*beta*dv * v_edge
        a_cur[q] = b_cur[q] = c_prev[q] = y_prev[q] = 0.0f;
    }

    // ---------------- forward sweep ----------------
    for (int jt = 0; jt < NVV; jt += TJ) {
        // async-stage f tile (each warp stages ROWS/2 = 64 rows)
#pragma unroll 4
        for (int rr = 0; rr < ROWS / 2; ++rr) {
            const int rl = wrp * (ROWS / 2) + rr;
            async_ld32(&f[(size_t)(row0 + rl) * NVV + jt + lane],
                       &tile[rl][lane]);
        }
        wait_async0();
        __syncthreads();

        for (int jj = 0; jj < TJ; ++jj) {
            const int j = jt + jj;
            const float vj   = (j + 0.5f) * DVF;
            const float coef = dt / (vj * vj * DVF);
            const float vE   = (j + 1) * DVF;
            const bool  last = (j == NVV - 1);
#pragma unroll
            for (int q = 0; q < RPT; ++q) {
                float a_next, b_next;
                if (last) {
                    a_next = 0.0f; b_next = 0.0f;
                } else {
                    const float w = tbdv[q] * vE;
                    const float delta = (w < 1e-8f)
                        ? 0.5f
                        : (__builtin_amdgcn_rcpf(w) -
                           __builtin_amdgcn_rcpf(expm1f(w)));
                    a_next = vE * delta - DoDV[q];
                    b_next = vE - vE * delta + DoDV[q];
                }
                const float l = coef * a_cur[q];
                const float d = 1.0f - coef * (a_next - b_cur[q]);
                const float u = -coef * b_next;
                const float r = tile[q * BTHREADS + tid][jj];
                const float denom = (j == 0) ? d : fmaf(-l, c_prev[q], d);
                const float rden  = __builtin_amdgcn_rcpf(denom);
                const float c = u * rden;
                const float y = ((j == 0) ? r : fmaf(-l, y_prev[q], r)) * rden;
                c_ws[(size_t)j * NXX + rowq[q]] = c;      // coalesced (transposed)
                ytile[q * BTHREADS + tid][jj] = y;
                c_prev[q] = c; y_prev[q] = y;
                a_cur[q] = a_next; b_cur[q] = b_next;
            }
        }
        __syncthreads();
        // async-stage y tile out to d_out (parked; overwritten by x later)
#pragma unroll 4
        for (int rr = 0; rr < ROWS / 2; ++rr) {
            const int rl = wrp * (ROWS / 2) + rr;
            async_st32(&out[(size_t)(row0 + rl) * NVV + jt + lane],
                       &ytile[rl][lane]);
        }
        wait_async0();      // ytile reused next iteration
        __syncthreads();
    }

    // ---------------- backward substitution ----------------
    float x_next[RPT];
#pragma unroll
    for (int q = 0; q < RPT; ++q) x_next[q] = 0.0f;

    for (int jt = NVV - TJ; jt >= 0; jt -= TJ) {
        // async-stage y (from out) and c tiles
#pragma unroll 4
        for (int rr = 0; rr < ROWS / 2; ++rr) {
            const int rl = wrp * (ROWS / 2) + rr;
            async_ld32(&out[(size_t)(row0 + rl) * NVV + jt + lane],
                       &ytile[rl][lane]);
        }
        // c tile: linear cooperative scheme, coalesced (j fixed per warp step)
#pragma unroll 2
        for (int i = 0; i < (ROWS * TJ) / BTHREADS; ++i) {
            const int e    = i * BTHREADS + tid;
            const int jjq  = e >> 7;           // / ROWS (=128)
            const int rloc = e & (ROWS - 1);
            async_ld32(&c_ws[(size_t)(jt + jjq) * NXX + row0 + rloc],
                       &ctile[rloc][jjq]);
        }
        wait_async0();
        __syncthreads();

        for (int jj = TJ - 1; jj >= 0; --jj) {
            const int j = jt + jj;
#pragma unroll
            for (int q = 0; q < RPT; ++q) {
                const float y = ytile[q * BTHREADS + tid][jj];
                float x;
                if (j == NVV - 1) x = y;
                else x = fmaf(-ctile[q * BTHREADS + tid][jj], x_next[q], y);
                tile[q * BTHREADS + tid][jj] = x;
                x_next[q] = x;
            }
        }
        __syncthreads();
        // async-stage x tile to d_out
#pragma unroll 4
        for (int rr = 0; rr < ROWS / 2; ++rr) {
            const int rl = wrp * (ROWS / 2) + rr;
            async_st32(&out[(size_t)(row0 + rl) * NVV + jt + lane],
                       &tile[rl][lane]);
        }
        wait_async0();      // tile reused next iteration (and before endpgm)
        __syncthreads();
    }
}

// ---------------------------------------------------------------------------
extern "C" void kernel_launch(void* const* d_in, const int* in_sizes, int n_in,
                              void* d_out, int out_size, void* d_ws, size_t ws_size,
                              hipStream_t stream)
{
    (void)in_sizes; (void)n_in; (void)out_size; (void)ws_size;
    // setup_inputs order: nu (scalar, unused: NUEE const), f0x, dt (scalar)
    const float* f0x = (const float*)d_in[1];
    const float* dtp = (const float*)d_in[2];
    float* out = (float*)d_out;

    float* wsf  = (float*)d_ws;
    float* beta = wsf;            // NXX floats
    float* c_ws = wsf + NXX;      // NXX*NVV floats (~16 MB)

    hipLaunchKernelGGL(moments_beta_kernel, dim3(NXX / 16), dim3(32), 0, stream,
                       f0x, beta);
    hipLaunchKernelGGL(thomas_kernel, dim3(NXX / ROWS), dim3(BTHREADS), 0, stream,
                       f0x, beta, dtp, c_ws, out);
}